// UMABackbone_25202868093027
// MI455X (gfx1250) — compile-verified
//
#include <hip/hip_runtime.h>
#include <hip/hip_bf16.h>
#include <math.h>

// ---------------------------------------------------------------------------
// UMA-backbone forward for MI455X (gfx1250, wave32).
// Memory-bound (~1.2 GB HBM traffic @ 23.3 TB/s vs ~125 GFLOP fp32): fp32 WMMA
// (V_WMMA_F32_16X16X4_F32) keeps reference precision at zero bandwidth cost.
// GEMM: one wave per 16x128 output tile; W k-panels are staged into LDS with
// the gfx1250 async global->LDS path (double buffered), A rows via b64 loads.
// ---------------------------------------------------------------------------

typedef __attribute__((ext_vector_type(2))) float v2f;
typedef __attribute__((ext_vector_type(8))) float v8f;

#define XE_DIM 768   // NB(512) + 2*EC(128)
#define GKC    32    // k-panel depth staged in LDS
#define GPAD   136   // padded LDS row stride (floats): 2 rows = 272 ≡ 16 mod 64 banks

#if defined(__AMDGCN__) && __has_builtin(__builtin_amdgcn_global_load_async_to_lds_b128)
#define HAS_ASYNC_LDS 1
#else
#define HAS_ASYNC_LDS 0
#endif

#if HAS_ASYNC_LDS
typedef int async_v4i __attribute__((vector_size(16)));
typedef __attribute__((address_space(1))) async_v4i* g_v4i_ptr;
typedef __attribute__((address_space(3))) async_v4i* l_v4i_ptr;
#endif

__device__ __forceinline__ float dsilu(float x)    { return x / (1.f + __expf(-x)); }
__device__ __forceinline__ float dsigmoid(float x) { return 1.f / (1.f + __expf(-x)); }

__device__ __forceinline__ void wait_async_lds()
{
#if HAS_ASYNC_LDS
#if __has_builtin(__builtin_amdgcn_s_wait_asynccnt)
    __builtin_amdgcn_s_wait_asynccnt(0);
#else
    asm volatile("s_wait_asynccnt 0" ::: "memory");
#endif
#endif
}

// ===========================================================================
// Generic WMMA GEMM:  Y[r, 0:128] (+)= sign * X[r, 0:K] @ W[0:K, 0:128] (+bias)
// Optional row LayerNorm over the 128 output cols, then SiLU / sigmoid.
// Block = 256 threads = 8 waves; each wave owns one 16-row x 128-col tile.
// W k-panels (GKC x 128) are staged in LDS (async, double buffered) and shared
// by all 8 waves. Fragment layouts per CDNA5 ISA:
//   A(16x4 f32):  lane = m + 16*(k/2), vgpr = k%2
//   B(4x16 f32):  lane = n + 16*(k/2), vgpr = k%2
//   C/D(16x16):   lane = n + 16*(m/8), vgpr = m%8
// K must be a multiple of GKC (true for 768/384/256/128).
// ===========================================================================
__global__ __launch_bounds__(256) void gemm128_wmma(
    const float* __restrict__ X, int ldx,
    const float* __restrict__ W, int ldw,
    const float* __restrict__ bias,
    const float* __restrict__ lng, const float* __restrict__ lnb,
    float* __restrict__ Y, int ldy,
    int M, int K, float sign, int accum, int act, int do_ln)
{
    __shared__ float sW[2][GKC][GPAD];

    const int tid    = threadIdx.x;
    const int waveId = tid >> 5;
    const int lane   = tid & 31;
    const int row0   = (blockIdx.x * 8 + waveId) << 4;

    const int lo = lane & 15;   // A: row m ; B/C: col n
    const int hi = lane >> 4;   // k-pair half / row half

    // --- cooperative staging of one GKC x 128 W panel into LDS -------------
    auto stage = [&](int buf, int kbase) {
#pragma unroll
        for (int i = 0; i < (GKC * 128 / 4) / 256; ++i) {   // 4 chunks of 16B
            int q  = tid + 256 * i;
            int k  = q >> 5;
            int c4 = (q & 31) << 2;
            const float* g = W + (size_t)(kbase + k) * (size_t)ldw + c4;
            float*       l = &sW[buf][k][c4];
#if HAS_ASYNC_LDS
            __builtin_amdgcn_global_load_async_to_lds_b128(
                (g_v4i_ptr)(void*)const_cast<float*>(g),
                (l_v4i_ptr)(void*)l, 0, 0);
#else
            *(float4*)l = *(const float4*)g;
#endif
        }
    };

    v8f acc[8];
#pragma unroll
    for (int t = 0; t < 8; ++t)
#pragma unroll
        for (int q = 0; q < 8; ++q) acc[t][q] = 0.f;

    const int  arow = row0 + lo;
    const bool av   = (arow < M);
    const float* xrow = X + (size_t)(av ? arow : 0) * (size_t)ldx + 2 * hi;

    const int nP = K / GKC;
    stage(0, 0);
    wait_async_lds();
    __syncthreads();

    for (int p = 0; p < nP; ++p) {
        if (p + 1 < nP) stage((p + 1) & 1, (p + 1) * GKC);

        const float (*panel)[GPAD] = sW[p & 1];

        v2f a[GKC / 4];
#pragma unroll
        for (int s = 0; s < GKC / 4; ++s) {                 // contiguous b64 loads
            float a0 = xrow[p * GKC + 4 * s];
            float a1 = xrow[p * GKC + 4 * s + 1];
            a[s].x = av ? a0 : 0.f;
            a[s].y = av ? a1 : 0.f;
        }
#pragma unroll
        for (int s = 0; s < GKC / 4; ++s) {
            const int kk = 4 * s + 2 * hi;
            const float* bp = &panel[kk][lo];
#pragma unroll
            for (int t = 0; t < 8; ++t) {
                v2f b;
                b.x = bp[t * 16];            // W[kk  ][t*16+lo]
                b.y = bp[GPAD + t * 16];     // W[kk+1][t*16+lo]
                acc[t] = __builtin_amdgcn_wmma_f32_16x16x4_f32(
                    false, a[s], false, b, (short)0, acc[t], false, false);
            }
        }
        wait_async_lds();
        __syncthreads();
    }

    // --- epilogue ----------------------------------------------------------
    if (sign != 1.f) {
#pragma unroll
        for (int t = 0; t < 8; ++t)
#pragma unroll
            for (int q = 0; q < 8; ++q) acc[t][q] *= sign;
    }
    if (bias) {
        float bv[8];
#pragma unroll
        for (int t = 0; t < 8; ++t) bv[t] = bias[t * 16 + lo];
#pragma unroll
        for (int t = 0; t < 8; ++t)
#pragma unroll
            for (int q = 0; q < 8; ++q) acc[t][q] += bv[t];
    }
    if (do_ln) {
        float gv[8], bev[8];
#pragma unroll
        for (int t = 0; t < 8; ++t) { gv[t] = lng[t * 16 + lo]; bev[t] = lnb[t * 16 + lo]; }
#pragma unroll
        for (int r = 0; r < 8; ++r) {          // one output row per r (within half)
            float s = 0.f, sq = 0.f;
#pragma unroll
            for (int t = 0; t < 8; ++t) { float v = acc[t][r]; s += v; sq += v * v; }
#pragma unroll
            for (int d = 1; d < 16; d <<= 1) { // reduce across the 16-lane half
                s  += __shfl_xor(s,  d, 32);
                sq += __shfl_xor(sq, d, 32);
            }
            float mu  = s * (1.f / 128.f);
            float var = sq * (1.f / 128.f) - mu * mu;
            float rs  = rsqrtf(var + 1e-5f);
#pragma unroll
            for (int t = 0; t < 8; ++t)
                acc[t][r] = (acc[t][r] - mu) * rs * gv[t] + bev[t];
        }
    }
    if (act == 1) {
#pragma unroll
        for (int t = 0; t < 8; ++t)
#pragma unroll
            for (int q = 0; q < 8; ++q) acc[t][q] = dsilu(acc[t][q]);
    } else if (act == 2) {
#pragma unroll
        for (int t = 0; t < 8; ++t)
#pragma unroll
            for (int q = 0; q < 8; ++q) acc[t][q] = dsigmoid(acc[t][q]);
    }

#pragma unroll
    for (int r = 0; r < 8; ++r) {
        int row = row0 + 8 * hi + r;
        if (row < M) {
            float* yp = Y + (size_t)row * (size_t)ldy + lo;
#pragma unroll
            for (int t = 0; t < 8; ++t) {
                if (accum) yp[t * 16] += acc[t][r];
                else       yp[t * 16]  = acc[t][r];
            }
        }
    }
}

// ===========================================================================
// Small per-system / per-edge / per-atom kernels
// ===========================================================================
__global__ __launch_bounds__(128) void sysemb_kernel(
    const float* __restrict__ charge, const float* __restrict__ spin,
    const float* __restrict__ Wcsd, const float* __restrict__ bcsd,
    float* __restrict__ sysemb)
{
    int s = blockIdx.x, c = threadIdx.x;
    __shared__ float csd[256];
    float ch = charge[s], sp = spin[s];
    const float lscale = logf(10000.f) / 64.f;
    for (int i = c; i < 256; i += 128) {
        int grp = i >> 6, f = i & 63;
        float freq = __expf(-(float)f * lscale);
        float ang  = ((grp < 2) ? ch : sp) * freq;
        csd[i] = (grp & 1) ? cosf(ang) : sinf(ang);
    }
    __syncthreads();
    float acc = bcsd[c];
    for (int i = 0; i < 256; ++i) acc += csd[i] * Wcsd[i * 128 + c];
    sysemb[s * 128 + c] = dsilu(acc);
}

__constant__ float cJ1[9]  = {0,1,0, 1,0,0, 0,0,-1};
__constant__ float cJ2[25] = {0,0,0,1,0,
                              0,1,0,0,0,
                              0,0,-0.5f,0,-0.8660254037844386f,
                              1,0,0,0,0,
                              0,0,-0.8660254037844386f,0,0.5f};
__constant__ int   cPerm[9] = {0,2,6,3,7,1,5,8,4};

__device__ float wig_entry(int srow, int col, float alpha, float beta)
{
    int l, a;
    if (srow == 0)      { l = 0; a = 0; }
    else if (srow < 4)  { l = 1; a = 1; }
    else                { l = 2; a = 4; }
    int lc;
    if (col == 0)       { lc = 0; }
    else if (col < 4)   { lc = 1; }
    else                { lc = 2; }
    if (l != lc) return 0.f;
    if (l == 0)  return 1.f;
    int n = 2 * l + 1;
    int i = srow - a, jj = col - a;
    const float* J = (l == 1) ? cJ1 : cJ2;
    // A(k,jj) = row k of (J @ Zb @ J) at column jj
    auto Afun = [&](int k) {
        float s = 0.f;
        for (int p = 0; p < n; ++p) {
            float fb = (float)(l - p);
            float b1 = cosf(fb * beta) * J[p * n + jj];
            if (p != n - 1 - p) b1 += sinf(fb * beta) * J[(n - 1 - p) * n + jj];
            s += J[k * n + p] * b1;
        }
        return s;
    };
    float fa = (float)(l - i);
    float v  = cosf(fa * alpha) * Afun(i);
    if (i != n - 1 - i) v += sinf(fa * alpha) * Afun(n - 1 - i);
    return v;
}

__global__ __launch_bounds__(128) void edge_geom_kernel(
    const float* __restrict__ evec, const int* __restrict__ zat,
    const int* __restrict__ eidx,
    const float* __restrict__ src_emb, const float* __restrict__ tgt_emb,
    float* __restrict__ xedge, float* __restrict__ wigM, float* __restrict__ env,
    int E)
{
    int e = blockIdx.x, tid = threadIdx.x;
    float vx = evec[e * 3 + 0], vy = evec[e * 3 + 1], vz = evec[e * 3 + 2];
    float d  = sqrtf(vx * vx + vy * vy + vz * vz);
    float dm = fmaxf(d, 1e-8f);
    float ny = vy / dm;
    float alpha = atan2f(vx / dm, vz / dm);
    float beta  = acosf(fminf(fmaxf(ny, -1.f), 1.f));
    if (tid == 0) {
        float u = d * 0.2f, ev = 0.f;
        if (u < 1.f) {
            float u5 = u * u * u * u * u;
            ev = 1.f - 21.f * u5 + 35.f * u5 * u - 15.f * u5 * u * u;
        }
        env[e] = ev;
    }
    int zs = zat[eidx[e]];
    int zt = zat[eidx[E + e]];
    const float step  = 5.f / 511.f;
    const float kcoef = -0.5f / ((step * 2.f) * (step * 2.f));
    for (int i = tid; i < XE_DIM; i += 128) {
        float val;
        if (i < 512)      { float t = d - step * (float)i; val = __expf(kcoef * t * t); }
        else if (i < 640) { val = src_emb[zs * 128 + (i - 512)] - 0.001f; }
        else              { val = tgt_emb[zt * 128 + (i - 640)] - 0.001f; }
        xedge[(size_t)e * XE_DIM + i] = val;
    }
    if (tid < 81) {
        int r = tid / 9, j = tid % 9;
        wigM[(size_t)e * 81 + r * 9 + j] = wig_entry(cPerm[r], j, alpha, beta);
    }
}

__global__ void init_x_kernel(const float* __restrict__ sphere_emb,
                              const float* __restrict__ sysemb,
                              const int* __restrict__ z, const int* __restrict__ batch,
                              float* __restrict__ x, int N)
{
    int i = blockIdx.x * blockDim.x + threadIdx.x;
    if (i >= N * 1152) return;
    int a = i / 1152, r = i % 1152, j = r >> 7, c = r & 127;
    float v = 0.f;
    if (j == 0) v = sphere_emb[z[a] * 128 + c] + sysemb[batch[a] * 128 + c];
    x[i] = v;
}

__global__ void zero_kernel(float* __restrict__ p, long long n)
{
    long long i = (long long)blockIdx.x * blockDim.x + threadIdx.x;
    long long s = (long long)gridDim.x * blockDim.x;
    for (; i < n; i += s) p[i] = 0.f;
}

// dst[tgt, j, c] += env*scale * sum_m wigM[e, m, j] * src[e, m, c]
template <int MC>
__global__ __launch_bounds__(128) void scatter_rot_kernel(
    const float* __restrict__ src, int srcStride,
    const float* __restrict__ wigM, const float* __restrict__ env,
    const int* __restrict__ eidx, int E, float scale,
    float* __restrict__ dst)
{
    int e = blockIdx.x, c = threadIdx.x;
    __shared__ float wm[81];
    if (c < 81) wm[c] = wigM[(size_t)e * 81 + c];
    __syncthreads();
    int tgt = eidx[E + e];
    float sc = env[e] * scale;
    float vals[MC];
#pragma unroll
    for (int m = 0; m < MC; ++m) vals[m] = src[(size_t)e * srcStride + m * 128 + c];
#pragma unroll
    for (int j = 0; j < 9; ++j) {
        float s = 0.f;
#pragma unroll
        for (int m = 0; m < MC; ++m) s += wm[m * 9 + j] * vals[m];
        atomicAdd(&dst[(size_t)tgt * 1152 + j * 128 + c], s * sc);
    }
}

__global__ __launch_bounds__(128) void rmsnorm_kernel(
    const float* __restrict__ x, const float* __restrict__ w,
    const float* __restrict__ sysemb, const int* __restrict__ batch,
    float* __restrict__ out)
{
    int a = blockIdx.x, c = threadIdx.x;
    float v[9];
#pragma unroll
    for (int j = 0; j < 9; ++j) v[j] = x[(size_t)a * 1152 + j * 128 + c];
    __shared__ float sb[384];
    sb[c]       = v[0] * v[0];
    sb[128 + c] = v[1] * v[1] + v[2] * v[2] + v[3] * v[3];
    sb[256 + c] = v[4] * v[4] + v[5] * v[5] + v[6] * v[6] + v[7] * v[7] + v[8] * v[8];
    __syncthreads();
    for (int s = 64; s > 0; s >>= 1) {
        if (c < s) {
            sb[c]       += sb[c + s];
            sb[128 + c] += sb[128 + c + s];
            sb[256 + c] += sb[256 + c + s];
        }
        __syncthreads();
    }
    float n0 = rsqrtf(sb[0]   * (1.f / 128.f) + 1e-6f);
    float n1 = rsqrtf(sb[128] * (1.f / 384.f) + 1e-6f);
    float n2 = rsqrtf(sb[256] * (1.f / 640.f) + 1e-6f);
    float o0 = v[0] * n0 * w[c];
    if (sysemb) o0 += sysemb[batch[a] * 128 + c];
    out[(size_t)a * 1152 + c] = o0;
#pragma unroll
    for (int j = 1; j < 4; ++j) out[(size_t)a * 1152 + j * 128 + c] = v[j] * n1 * w[128 + c];
#pragma unroll
    for (int j = 4; j < 9; ++j) out[(size_t)a * 1152 + j * 128 + c] = v[j] * n2 * w[256 + c];
}

// xr[e,m,c] = einv[e,c] * sum_j wigM[e,m,j] * (xn[src,j,c] + xn[tgt,j,c])
__global__ __launch_bounds__(128) void message_rot_kernel(
    const float* __restrict__ xn, const float* __restrict__ wigM,
    const float* __restrict__ einv, const int* __restrict__ eidx,
    int E, float* __restrict__ xr)
{
    int e = blockIdx.x, c = threadIdx.x;
    __shared__ float wm[81];
    if (c < 81) wm[c] = wigM[(size_t)e * 81 + c];
    __syncthreads();
    int s = eidx[e], t = eidx[E + e];
    float msg[9];
#pragma unroll
    for (int j = 0; j < 9; ++j)
        msg[j] = xn[(size_t)s * 1152 + j * 128 + c] + xn[(size_t)t * 1152 + j * 128 + c];
    float ei = einv[(size_t)e * 128 + c];
#pragma unroll
    for (int m = 0; m < 9; ++m) {
        float acc = 0.f;
#pragma unroll
        for (int j = 0; j < 9; ++j) acc += wm[m * 9 + j] * msg[j];
        xr[(size_t)e * 1152 + m * 128 + c] = acc * ei;
    }
}

__global__ __launch_bounds__(128) void gate_apply_kernel(
    const float* __restrict__ agg, const float* __restrict__ gate,
    float* __restrict__ x)
{
    int a = blockIdx.x, c = threadIdx.x;
    float g0 = gate[(size_t)a * 256 + c];
    float g1 = gate[(size_t)a * 256 + 128 + c];
    float a0 = agg[(size_t)a * 1152 + c];
    x[(size_t)a * 1152 + c] += dsilu(a0);
#pragma unroll
    for (int j = 1; j < 4; ++j) x[(size_t)a * 1152 + j * 128 + c] += agg[(size_t)a * 1152 + j * 128 + c] * g0;
#pragma unroll
    for (int j = 4; j < 9; ++j) x[(size_t)a * 1152 + j * 128 + c] += agg[(size_t)a * 1152 + j * 128 + c] * g1;
}

__global__ __launch_bounds__(128) void to_grid_kernel(
    const float* __restrict__ xn2, const float* __restrict__ tg,
    float* __restrict__ xg)
{
    int a = blockIdx.x, c = threadIdx.x;
    __shared__ float sg[162];
    for (int i = c; i < 162; i += 128) sg[i] = tg[i];
    __syncthreads();
    float v[9];
#pragma unroll
    for (int j = 0; j < 9; ++j) v[j] = xn2[(size_t)a * 1152 + j * 128 + c];
#pragma unroll
    for (int g = 0; g < 18; ++g) {
        float s = 0.f;
#pragma unroll
        for (int j = 0; j < 9; ++j) s += sg[g * 9 + j] * v[j];
        xg[(size_t)a * 2304 + g * 128 + c] = s;
    }
}

__global__ __launch_bounds__(128) void from_grid_kernel(
    const float* __restrict__ xg2, const float* __restrict__ fg,
    float* __restrict__ x)
{
    int a = blockIdx.x, c = threadIdx.x;
    __shared__ float sf[162];
    for (int i = c; i < 162; i += 128) sf[i] = fg[i];
    __syncthreads();
    float v[18];
#pragma unroll
    for (int g = 0; g < 18; ++g) v[g] = xg2[(size_t)a * 2304 + g * 128 + c];
#pragma unroll
    for (int j = 0; j < 9; ++j) {
        float s = 0.f;
#pragma unroll
        for (int g = 0; g < 18; ++g) s += sf[j * 18 + g] * v[g];
        x[(size_t)a * 1152 + j * 128 + c] += s;
    }
}

// ===========================================================================
// Host orchestration
// ===========================================================================
struct RadP { const float *W1,*W2,*Wout,*b1,*b2,*be1,*be2,*bout,*g1,*g2; };

static inline void launch_gemm(hipStream_t st,
    const float* X, int ldx, const float* W, int ldw,
    const float* bias, const float* g, const float* be,
    float* Y, int ldy, int M, int K,
    float sign, int accum, int act, int ln)
{
    int tiles  = (M + 15) / 16;
    int blocks = (tiles + 7) / 8;
    gemm128_wmma<<<blocks, 256, 0, st>>>(X, ldx, W, ldw, bias, g, be, Y, ldy,
                                         M, K, sign, accum, act, ln);
}

static inline void run_radial(hipStream_t st, const float* xedge, const RadP& r,
                              int dout, float* h1, float* h2, float* out,
                              int outAct, int M)
{
    launch_gemm(st, xedge, XE_DIM, r.W1, 128, r.b1, r.g1, r.be1, h1, 128, M, XE_DIM, 1.f, 0, 1, 1);
    launch_gemm(st, h1,    128,    r.W2, 128, r.b2, r.g2, r.be2, h2, 128, M, 128,    1.f, 0, 1, 1);
    for (int p = 0; p < dout / 128; ++p)
        launch_gemm(st, h2, 128, r.Wout + p * 128, dout, r.bout + p * 128,
                    nullptr, nullptr, out + p * 128, dout, M, 128, 1.f, 0, outAct, 0);
}

// --- input index map -------------------------------------------------------
// top-level dict insertion order; 'params' flattened as a JAX pytree
// (dict keys sorted, lists in order).
enum {
    IN_POS = 0, IN_EVEC = 1, IN_CHARGE = 2, IN_SPIN = 3,
    IN_WCSD = 4, IN_BCSD = 5,
    IN_BLK0 = 6,            // 23 leaves per block
    IN_BLK1 = 29,
    IN_EDEG = 52,           // 10 leaves
    IN_FROM_GRID = 62, IN_NORM_OUT = 63, IN_SPHERE = 64,
    IN_SRC = 65, IN_TGT = 66, IN_TO_GRID = 67,
    IN_Z = 68, IN_BATCH = 69, IN_EIDX = 70
};
// block leaf offsets (sorted keys): W0,W1i,W1r,W2i,W2r,Wff1,Wff2,Wg,
//   bff1,bff2,bg,norm1,norm2, rad{W1,W2,Wout,b1,b2,be1,be2,bout,g1,g2}
enum {
    BL_W0 = 0, BL_W1I = 1, BL_W1R = 2, BL_W2I = 3, BL_W2R = 4,
    BL_WFF1 = 5, BL_WFF2 = 6, BL_WG = 7, BL_BFF1 = 8, BL_BFF2 = 9,
    BL_BG = 10, BL_NORM1 = 11, BL_NORM2 = 12, BL_RAD = 13
};

extern "C" void kernel_launch(void* const* d_in, const int* in_sizes, int n_in,
                              void* d_out, int out_size, void* d_ws, size_t ws_size,
                              hipStream_t stream)
{
    (void)n_in; (void)out_size; (void)ws_size;
    auto F = [&](int i) { return (const float*)d_in[i]; };
    auto I = [&](int i) { return (const int*)d_in[i]; };

    const int N    = in_sizes[IN_Z];            // 5000
    const int E    = in_sizes[IN_EIDX] / 2;     // 40000
    const int NSYS = in_sizes[IN_CHARGE];       // 64

    const int* zat   = I(IN_Z);
    const int* batch = I(IN_BATCH);
    const int* eidx  = I(IN_EIDX);

    // workspace carve-up (floats)
    float* w = (float*)d_ws;
    size_t off = 0;
    float* sysembW = w + off; off += (size_t)NSYS * 128;
    float* xedgeW  = w + off; off += (size_t)E * XE_DIM;
    float* wigMW   = w + off; off += (size_t)E * 81;
    float* envW    = w + off; off += (size_t)E;
    float* h1W     = w + off; off += (size_t)E * 128;
    float* h2W     = w + off; off += (size_t)E * 128;
    float* radoutW = w + off; off += (size_t)E * 384;   // also reused as xg (N*2304 < E*384)
    float* einvW   = w + off; off += (size_t)E * 128;
    float* xW      = w + off; off += (size_t)N * 1152;
    float* xnW     = w + off; off += (size_t)N * 1152;
    float* xrW     = w + off; off += (size_t)E * 1152;
    float* outbW   = w + off; off += (size_t)E * 1152;
    float* aggW    = w + off; off += (size_t)N * 1152;
    float* gateW   = w + off; off += (size_t)N * 256;
    float* xghW    = w + off; off += (size_t)N * 2304;
    float* xgW     = radoutW;   // alias: radout dead after initial embedding

    // 1) system embedding
    sysemb_kernel<<<NSYS, 128, 0, stream>>>(F(IN_CHARGE), F(IN_SPIN),
                                            F(IN_WCSD), F(IN_BCSD), sysembW);
    // 2) edge geometry: x_edge, Wigner (permuted), envelope
    edge_geom_kernel<<<E, 128, 0, stream>>>(F(IN_EVEC), zat, eidx,
                                            F(IN_SRC), F(IN_TGT),
                                            xedgeW, wigMW, envW, E);
    // 3) x init
    init_x_kernel<<<(N * 1152 + 255) / 256, 256, 0, stream>>>(
        F(IN_SPHERE), sysembW, zat, batch, xW, N);

    // 4) initial edge-degree embedding: w0 = radial(x_edge, edeg) (E,3,C)
    RadP edeg = { F(IN_EDEG+0), F(IN_EDEG+1), F(IN_EDEG+2), F(IN_EDEG+3), F(IN_EDEG+4),
                  F(IN_EDEG+5), F(IN_EDEG+6), F(IN_EDEG+7), F(IN_EDEG+8), F(IN_EDEG+9) };
    run_radial(stream, xedgeW, edeg, 384, h1W, h2W, radoutW, /*act*/0, E);
    // 5) x[tgt] += rot_inv(w0) * env / 5
    scatter_rot_kernel<3><<<E, 128, 0, stream>>>(radoutW, 384, wigMW, envW,
                                                 eidx, E, 0.2f, xW);

    // 6) message-passing blocks
    for (int b = 0; b < 2; ++b) {
        const int B = (b == 0) ? IN_BLK0 : IN_BLK1;
        const float *W0  = F(B+BL_W0),  *W1i = F(B+BL_W1I), *W1r = F(B+BL_W1R);
        const float *W2i = F(B+BL_W2I), *W2r = F(B+BL_W2R);
        RadP rad = { F(B+BL_RAD+0), F(B+BL_RAD+1), F(B+BL_RAD+2), F(B+BL_RAD+3),
                     F(B+BL_RAD+4), F(B+BL_RAD+5), F(B+BL_RAD+6), F(B+BL_RAD+7),
                     F(B+BL_RAD+8), F(B+BL_RAD+9) };

        // a) xn = rms_norm_sh(x, norm1); xn[:,0,:] += sysemb[batch]
        rmsnorm_kernel<<<N, 128, 0, stream>>>(xW, F(B+BL_NORM1), sysembW, batch, xnW);
        // b) e_inv = silu(radial(x_edge))
        run_radial(stream, xedgeW, rad, 128, h1W, h2W, einvW, /*act silu*/1, E);
        // c) xr = rot(xn[src]+xn[tgt]) * e_inv
        message_rot_kernel<<<E, 128, 0, stream>>>(xnW, wigMW, einvW, eidx, E, xrW);

        // d) channel mixes -> outb  (col layout: [m0(3) | o1r(2) | o1i(2) | o2r | o2i])
        for (int p = 0; p < 3; ++p)
            launch_gemm(stream, xrW, 1152, W0 + p*128, 384, nullptr, nullptr, nullptr,
                        outbW + p*128, 1152, E, 384, 1.f, 0, 0, 0);
        for (int p = 0; p < 2; ++p) {
            launch_gemm(stream, xrW+384, 1152, W1r + p*128, 256, nullptr, nullptr, nullptr,
                        outbW+384 + p*128, 1152, E, 256,  1.f, 0, 0, 0);
            launch_gemm(stream, xrW+640, 1152, W1i + p*128, 256, nullptr, nullptr, nullptr,
                        outbW+384 + p*128, 1152, E, 256, -1.f, 1, 0, 0);
            launch_gemm(stream, xrW+384, 1152, W1i + p*128, 256, nullptr, nullptr, nullptr,
                        outbW+640 + p*128, 1152, E, 256,  1.f, 0, 0, 0);
            launch_gemm(stream, xrW+640, 1152, W1r + p*128, 256, nullptr, nullptr, nullptr,
                        outbW+640 + p*128, 1152, E, 256,  1.f, 1, 0, 0);
        }
        launch_gemm(stream, xrW+896,  1152, W2r, 128, nullptr, nullptr, nullptr,
                    outbW+896,  1152, E, 128,  1.f, 0, 0, 0);
        launch_gemm(stream, xrW+1024, 1152, W2i, 128, nullptr, nullptr, nullptr,
                    outbW+896,  1152, E, 128, -1.f, 1, 0, 0);
        launch_gemm(stream, xrW+896,  1152, W2i, 128, nullptr, nullptr, nullptr,
                    outbW+1024, 1152, E, 128,  1.f, 0, 0, 0);
        launch_gemm(stream, xrW+1024, 1152, W2r, 128, nullptr, nullptr, nullptr,
                    outbW+1024, 1152, E, 128,  1.f, 1, 0, 0);

        // e/f) agg = scatter(rot_inv(out) * env)
        zero_kernel<<<2048, 256, 0, stream>>>(aggW, (long long)N * 1152);
        scatter_rot_kernel<9><<<E, 128, 0, stream>>>(outbW, 1152, wigMW, envW,
                                                     eidx, E, 1.f, aggW);
        // g) gate = sigmoid(agg[:,0,:] @ Wg + bg)
        for (int p = 0; p < 2; ++p)
            launch_gemm(stream, aggW, 1152, F(B+BL_WG) + p*128, 256, F(B+BL_BG) + p*128,
                        nullptr, nullptr, gateW + p*128, 256, N, 128, 1.f, 0, 2, 0);
        // h) x += gated agg
        gate_apply_kernel<<<N, 128, 0, stream>>>(aggW, gateW, xW);

        // i-m) grid FFN
        rmsnorm_kernel<<<N, 128, 0, stream>>>(xW, F(B+BL_NORM2), nullptr, nullptr, xnW);
        to_grid_kernel<<<N, 128, 0, stream>>>(xnW, F(IN_TO_GRID), xgW);
        launch_gemm(stream, xgW,  128, F(B+BL_WFF1), 128, F(B+BL_BFF1), nullptr, nullptr,
                    xghW, 128, N * 18, 128, 1.f, 0, 1, 0);
        launch_gemm(stream, xghW, 128, F(B+BL_WFF2), 128, F(B+BL_BFF2), nullptr, nullptr,
                    xgW,  128, N * 18, 128, 1.f, 0, 0, 0);
        from_grid_kernel<<<N, 128, 0, stream>>>(xgW, F(IN_FROM_GRID), xW);
    }

    // 7) final norm -> output
    rmsnorm_kernel<<<N, 128, 0, stream>>>(xW, F(IN_NORM_OUT), nullptr, nullptr,
                                          (float*)d_out);
}